// USR_EMB_23476291240225
// MI455X (gfx1250) — compile-verified
//
#include <hip/hip_runtime.h>
#include <hip/hip_bf16.h>

// USR_EMB gather for MI455X (gfx1250).
//
// out[row, :] = emb_usr[searchsorted(userlist, x[row]), :]
//   rows = BATCH*HIST = 819200, EMB = 64 fp32 (256 B per row).
//
// Bandwidth-bound: ~210 MB streaming stores + gather reads of a 25.6 MB
// table that fits the 192 MB L2 (each row re-read ~8x). Data path:
// global --(async B128)--> LDS --(async B128, non-temporal)--> global,
// tracked with ASYNCcnt (gfx1250 async-copy path): gathered rows never
// occupy VGPRs, one s_wait_asynccnt per wave, and NT stores keep the
// write-once output stream from evicting the embedding table out of L2.

#define ROWS_PER_BLOCK 16
#define EMB_DIM 64          // floats per row
#define BLOCK_THREADS 256   // ROWS_PER_BLOCK * 16 chunks/row

#define CPOL_RT 0           // regular temporal (loads: cache the table)
#define CPOL_STORE_NT 1     // TH[2:0]=1 -> TH_STORE_NT (streaming output)

typedef int v4i __attribute__((ext_vector_type(4)));

__global__ __launch_bounds__(BLOCK_THREADS) void usr_emb_gather_kernel(
    const int* __restrict__ x,         // [n_rows] user ids
    const int* __restrict__ userlist,  // [n_ul] sorted (-1, 0..N-1)
    int n_rows,
    int n_ul,
    const float* __restrict__ emb,     // [n_ul * EMB_DIM]
    float* __restrict__ out)           // [n_rows * EMB_DIM]
{
    __shared__ int sIdx[ROWS_PER_BLOCK];
    __shared__ __align__(16) float sBuf[ROWS_PER_BLOCK * EMB_DIM]; // 4 KB stage

    const int  t       = threadIdx.x;
    const long rowBase = (long)blockIdx.x * ROWS_PER_BLOCK;

    // Phase 1: one binary search per row (threads 0..15), result shared in LDS.
    // userlist (400 KB as int32) is fully L2-resident; upper tree levels hit L0.
    if (t < ROWS_PER_BLOCK) {
        const long row = rowBase + t;
        int idx = 0;
        if (row < n_rows) {
            const int key = x[row];
            int lo = 0, hi = n_ul;            // searchsorted 'left'
            while (lo < hi) {
                const int mid = (lo + hi) >> 1;
                if (userlist[mid] < key) lo = mid + 1;
                else                     hi = mid;
            }
            idx = lo;
        }
        sIdx[t] = idx;
    }
    __syncthreads();

    const int  r   = t >> 4;            // row within tile
    const int  c   = t & 15;            // 16-byte chunk within row
    const long row = rowBase + r;
    if (row >= n_rows) return;

    // Each thread moves one 16-byte chunk: global -> LDS -> global.
    // Wave loads touch 2 contiguous 256B rows; wave stores are 512B contiguous.
    const float* gsrc = emb + ((long)sIdx[r] * EMB_DIM + (c << 2));
    float*       gdst = out + (row * EMB_DIM + (c << 2));
    float*       slot = &sBuf[t << 2];

    __builtin_amdgcn_global_load_async_to_lds_b128(
        (v4i*)gsrc, (v4i*)slot, /*offset=*/0, /*cpol=*/CPOL_RT);
    __builtin_amdgcn_s_wait_asynccnt(0);   // our wave's gathered chunks are in LDS
    __builtin_amdgcn_global_store_async_from_lds_b128(
        (v4i*)gdst, (v4i*)slot, /*offset=*/0, /*cpol=*/CPOL_STORE_NT);
    // S_ENDPGM implicitly waits for ASYNCcnt==0, so the store completes
    // before the block's LDS is released.
}

extern "C" void kernel_launch(void* const* d_in, const int* in_sizes, int n_in,
                              void* d_out, int out_size, void* d_ws, size_t ws_size,
                              hipStream_t stream) {
    const int*   x        = (const int*)d_in[0];    // ids
    const int*   userlist = (const int*)d_in[1];    // sorted userlist
    const float* emb      = (const float*)d_in[2];  // embedding table
    float*       out      = (float*)d_out;

    const int n_rows = in_sizes[0];                 // BATCH*HIST = 819200
    const int n_ul   = in_sizes[1];                 // USR_SIZE+1 = 100001

    const int grid = (n_rows + ROWS_PER_BLOCK - 1) / ROWS_PER_BLOCK;
    usr_emb_gather_kernel<<<grid, BLOCK_THREADS, 0, stream>>>(
        x, userlist, n_rows, n_ul, emb, out);
}